// DeconvLayer_39410619908359
// MI455X (gfx1250) — compile-verified
//
#include <hip/hip_runtime.h>

typedef __attribute__((ext_vector_type(2))) float v2f;
typedef __attribute__((ext_vector_type(8))) float v8f;

// Types for the async global->LDS builtin: b128 moves one int4-vector per lane.
typedef int v4i __attribute__((vector_size(16)));
typedef __attribute__((address_space(1))) v4i gv4i;  // global (HIP AS1)
typedef __attribute__((address_space(3))) v4i lv4i;  // LDS    (HIP AS3)

#define BATCH 256
#define SEQ   16384
#define FILT  16
#define T     64
#define NBLK  (SEQ / T)          // 256 serial block steps
#define WAVES 4                  // one wave per 16-col N-tile of the T-block
#define NTHREADS (WAVES * 32)    // 128 threads, wave32
#define STRIPS (BATCH / 16)      // 16 workgroups

#if __has_builtin(__builtin_amdgcn_global_load_async_to_lds_b128)
#define HAVE_ASYNC_LDS 1
#else
#define HAVE_ASYNC_LDS 0
#endif

// Stage 8 consecutive floats (32B) from global into LDS for this thread.
__device__ __forceinline__ void stage_x8(const float* xs, float* ldst) {
#if HAVE_ASYNC_LDS
    __builtin_amdgcn_global_load_async_to_lds_b128(
        (gv4i*)const_cast<float*>(xs), (lv4i*)ldst, 0, 0);
    __builtin_amdgcn_global_load_async_to_lds_b128(
        (gv4i*)const_cast<float*>(xs + 4), (lv4i*)(ldst + 4), 0, 0);
#else
    float4 a = *(const float4*)(xs);
    float4 b = *(const float4*)(xs + 4);
    *(float4*)(ldst)     = a;
    *(float4*)(ldst + 4) = b;
#endif
}

__device__ __forceinline__ void wait_async0() {
#if __has_builtin(__builtin_amdgcn_s_wait_asynccnt)
    __builtin_amdgcn_s_wait_asynccnt(0);
#else
    asm volatile("s_wait_asynccnt 0x0" ::: "memory");
#endif
}

// Two chained IIR passes, block-recurrence form:
//   V = X*G^T + S*P^T   per [16 rows x T cols] block, carry S = last 16 v's.
__global__ __launch_bounds__(NTHREADS)
void iir2_wmma_kernel(const float* __restrict__ x,
                      const float* __restrict__ h,
                      float* __restrict__ y)
{
    __shared__ float Gsh[T][T];      // lower-triangular impulse-response matrix
    __shared__ float Psh[T][FILT];   // state-injection matrix P = G*U
    __shared__ float Xb[2][16][T];   // double-buffered staged input block
    __shared__ float Vb[16][T];      // pass-1 output block
    __shared__ float V2b[16][T];     // pass-2 output block
    __shared__ float S1[16][FILT];   // pass-1 carry state (v[t-1-j])
    __shared__ float S2[16][FILT];   // pass-2 carry state
    __shared__ float gimp[T];        // IIR impulse response
    __shared__ float hsh[FILT];

    const int tid    = threadIdx.x;
    const int wave   = tid >> 5;          // N-tile index 0..3
    const int lane   = tid & 31;
    const int laneLo = lane & 15;
    const int laneHi = lane >> 4;         // 0/1
    const int row0   = blockIdx.x * 16;   // batch-row strip

    // ---- one-time setup: build g, G, P in LDS ----
    if (tid < FILT) hsh[tid] = h[tid];
    __syncthreads();

    if (tid == 0) {
        gimp[0] = 1.0f;
        for (int n = 1; n < T; ++n) {
            float s = 0.0f;
            int jmax = (n - 1 < FILT - 1) ? (n - 1) : (FILT - 1);
            for (int j = 0; j <= jmax; ++j) s += hsh[j] * gimp[n - 1 - j];
            gimp[n] = s;
        }
    }
    __syncthreads();

    for (int e = tid; e < T * T; e += NTHREADS) {
        int k = e >> 6, m = e & (T - 1);
        Gsh[k][m] = (k >= m) ? gimp[k - m] : 0.0f;
    }
    for (int e = tid; e < T * FILT; e += NTHREADS) {
        int k = e >> 4, j = e & 15;
        float s = 0.0f;
        int mmax = (k < FILT - 1 - j) ? k : (FILT - 1 - j);
        for (int m = 0; m <= mmax; ++m) s += gimp[k - m] * hsh[m + j];
        Psh[k][j] = s;
    }
    for (int e = tid; e < 16 * FILT; e += NTHREADS) {
        (&S1[0][0])[e] = 0.0f;
        (&S2[0][0])[e] = 0.0f;
    }
    __syncthreads();

    // ---- preload loop-invariant B fragments into registers ----
    // B tile for N-tile `wave`, K-step m:  B[kk][nn] = G[16*wave+nn][4m+kk]
    v2f Gfrag[T / 4];
#pragma unroll
    for (int m = 0; m < T / 4; ++m) {
        int c0 = 4 * m + 2 * laneHi;
        Gfrag[m].x = Gsh[16 * wave + laneLo][c0];
        Gfrag[m].y = Gsh[16 * wave + laneLo][c0 + 1];
    }
    v2f Pfrag[FILT / 4];
#pragma unroll
    for (int j = 0; j < FILT / 4; ++j) {
        int c0 = 4 * j + 2 * laneHi;
        Pfrag[j].x = Psh[16 * wave + laneLo][c0];
        Pfrag[j].y = Psh[16 * wave + laneLo][c0 + 1];
    }

    // cooperative load/store mapping: each thread moves 8 consecutive floats
    const int lr = tid >> 3;        // row 0..15
    const int lc = (tid & 7) * 8;   // col 0,8,..,56
    const float* xrow = x + (size_t)(row0 + lr) * SEQ + lc;
    float*       yrow = y + (size_t)(row0 + lr) * SEQ + lc;

    // ---- prologue: fetch block 0, apply reference's v[i<16]=0 rule ----
    stage_x8(xrow, &Xb[0][lr][lc]);
    wait_async0();
    if (lc < FILT) {
#pragma unroll
        for (int i = 0; i < 8; ++i) Xb[0][lr][lc + i] = 0.0f;
    }
    __syncthreads();

    int buf = 0;
    for (int blk = 0; blk < NBLK; ++blk) {
        const int t0 = blk * T;

        // kick off async fetch of the NEXT block into the other buffer;
        // it overlaps the entire WMMA body below (off the critical path)
        if (blk + 1 < NBLK)
            stage_x8(xrow + t0 + T, &Xb[buf ^ 1][lr][lc]);

        // ---- pass 1: V = X*G^T + S1*P^T ----
        v8f acc = {};
#pragma unroll
        for (int m = 0; m < T / 4; ++m) {
            v2f a;
            int c0 = 4 * m + 2 * laneHi;
            a.x = Xb[buf][laneLo][c0];
            a.y = Xb[buf][laneLo][c0 + 1];
            acc = __builtin_amdgcn_wmma_f32_16x16x4_f32(
                false, a, false, Gfrag[m], (short)0, acc, false, false);
        }
#pragma unroll
        for (int j = 0; j < FILT / 4; ++j) {
            v2f a;
            int c0 = 4 * j + 2 * laneHi;
            a.x = S1[laneLo][c0];
            a.y = S1[laneLo][c0 + 1];
            acc = __builtin_amdgcn_wmma_f32_16x16x4_f32(
                false, a, false, Pfrag[j], (short)0, acc, false, false);
        }
#pragma unroll
        for (int r = 0; r < 8; ++r)
            Vb[r + 8 * laneHi][16 * wave + laneLo] = acc[r];
        __syncthreads();

        // ---- pass 2: Y = V*G^T + S2*P^T ----
        v8f acc2 = {};
#pragma unroll
        for (int m = 0; m < T / 4; ++m) {
            v2f a;
            int c0 = 4 * m + 2 * laneHi;
            a.x = Vb[laneLo][c0];
            a.y = Vb[laneLo][c0 + 1];
            acc2 = __builtin_amdgcn_wmma_f32_16x16x4_f32(
                false, a, false, Gfrag[m], (short)0, acc2, false, false);
        }
#pragma unroll
        for (int j = 0; j < FILT / 4; ++j) {
            v2f a;
            int c0 = 4 * j + 2 * laneHi;
            a.x = S2[laneLo][c0];
            a.y = S2[laneLo][c0 + 1];
            acc2 = __builtin_amdgcn_wmma_f32_16x16x4_f32(
                false, a, false, Pfrag[j], (short)0, acc2, false, false);
        }
#pragma unroll
        for (int r = 0; r < 8; ++r)
            V2b[r + 8 * laneHi][16 * wave + laneLo] = acc2[r];

        // update pass-1 carry from Vb's last 16 columns (pass-1 reads done)
        {
            int e0 = tid * 2;
#pragma unroll
            for (int i = 0; i < 2; ++i) {
                int e = e0 + i, b = e >> 4, j = e & 15;
                S1[b][j] = Vb[b][T - 1 - j];
            }
        }
        __syncthreads();

        // write pass-2 block to global, update pass-2 carry
        float4 o1 = *(float4*)&V2b[lr][lc];
        float4 o2 = *(float4*)&V2b[lr][lc + 4];
        *(float4*)(yrow + t0)     = o1;
        *(float4*)(yrow + t0 + 4) = o2;
        {
            int e0 = tid * 2;
#pragma unroll
            for (int i = 0; i < 2; ++i) {
                int e = e0 + i, b = e >> 4, j = e & 15;
                S2[b][j] = V2b[b][T - 1 - j];
            }
        }

        // publish next X buffer: wait for its async copy, then barrier
        wait_async0();
        __syncthreads();
        buf ^= 1;
    }
}

extern "C" void kernel_launch(void* const* d_in, const int* in_sizes, int n_in,
                              void* d_out, int out_size, void* d_ws, size_t ws_size,
                              hipStream_t stream) {
    const float* x = (const float*)d_in[0];   // [256, 16384] fp32
    const float* h = (const float*)d_in[1];   // [1, 16] fp32
    float* y = (float*)d_out;                 // [256, 16384] fp32
    (void)in_sizes; (void)n_in; (void)out_size; (void)d_ws; (void)ws_size;
    iir2_wmma_kernel<<<dim3(STRIPS), dim3(NTHREADS), 0, stream>>>(x, h, y);
}